// FlowMatchingLoss_29016799051776
// MI455X (gfx1250) — compile-verified
//
#include <hip/hip_runtime.h>
#include <hip/hip_bf16.h>
#include <stdint.h>

typedef float v2f __attribute__((ext_vector_type(2)));
typedef float v8f __attribute__((ext_vector_type(8)));
typedef int   v4i __attribute__((ext_vector_type(4)));

#define BB   4
#define NN   4096
#define MM   16
#define TOPK 5
#define NDIV 100

#define AS1 __attribute__((address_space(1)))
#define AS3 __attribute__((address_space(3)))

// ---------------- block reduction (wave32 shuffle + LDS) ----------------
__device__ __forceinline__ float blk_reduce(float v, float* red) {
    #pragma unroll
    for (int o = 16; o > 0; o >>= 1) v += __shfl_down(v, o, 32);
    const int wv = threadIdx.x >> 5;
    const int ln = threadIdx.x & 31;
    __syncthreads();                 // protect `red` across consecutive calls
    if (ln == 0) red[wv] = v;
    __syncthreads();
    float s = 0.0f;
    if (threadIdx.x == 0) {
        const int nw = (blockDim.x + 31) >> 5;
        for (int i = 0; i < nw; ++i) s += red[i];
    }
    return s;                        // valid on thread 0 only
}

__device__ __forceinline__ uint32_t hashu(uint32_t x) {
    x ^= x >> 16; x *= 0x7feb352du;
    x ^= x >> 15; x *= 0x846ca68bu;
    x ^= x >> 16; return x;
}

// ---------------- velocity MSE: sum of (pred-targ)^2 -> ws[0] ----------------
__global__ __launch_bounds__(256) void vel_kernel(const float* __restrict__ p,
                                                  const float* __restrict__ t,
                                                  float* __restrict__ ws) {
    __shared__ float red[8];
    float acc = 0.0f;
    for (int i = blockIdx.x * blockDim.x + threadIdx.x; i < BB * NN * 2;
         i += gridDim.x * blockDim.x) {
        float d = p[i] - t[i];
        acc += d * d;
    }
    float s = blk_reduce(acc, red);
    if (threadIdx.x == 0) atomicAdd(&ws[0], s);
}

// ---------------- consistency loss via f32 WMMA distance tiles ----------------
// One wave per 16-row strip. D = A x B where
//   A row i = [x_i, y_i, s_i, 1],  B col j = [-2x_j, -2y_j, 1, s_j]
// gives D[i][j] = s_i + s_j - 2 p_i.p_j = squared distance (exact, K=4 WMMA).
// All tile/merge LDS exchange is intra-wave: CDNA5 LDS ops from one wave are
// processed in order, so only a compiler scheduling fence is needed (no block
// barrier). Positions for the block's batch are staged into LDS once, using
// the gfx1250 async-to-LDS path when the builtin is available.
__global__ __launch_bounds__(256) void cons_kernel(const float* __restrict__ pos,
                                                   const float* __restrict__ vel,
                                                   float* __restrict__ ws) {
    __shared__ float4 spos4[(NN * 2) / 4];              // 32 KB: pos[b] staged
    __shared__ __align__(16) float tile[8][16][20];     // per-wave D tile, padded
    __shared__ float md[8][2][16][TOPK];                // per-wave half top-K d^2
    __shared__ int   mi[8][2][16][TOPK];                // per-wave half top-K idx
    __shared__ float red[8];

    const int wv = threadIdx.x >> 5;           // wave in block (0..7)
    const int ln = threadIdx.x & 31;           // lane (wave32)
    const int h  = ln >> 4;                    // half: K-group select
    const int r  = ln & 15;                    // row (A) / col (B) within tile

    const int strip = blockIdx.x * 8 + wv;     // 16-row strip id
    const int b  = strip / (NN / 16);          // uniform across the block
    const int i0 = (strip % (NN / 16)) * 16;
    const float* posb = pos + (size_t)b * NN * 2;
    const float* velb = vel + (size_t)b * NN * 2;

    // ---- stage pos[b] (8192 floats) into LDS, async path if available ----
    {
        const float4* src = (const float4*)posb;
#if __has_builtin(__builtin_amdgcn_global_load_async_to_lds_b128)
        #pragma unroll
        for (int k = 0; k < (NN * 2 / 4) / 256; ++k) {
            const int e = threadIdx.x + k * 256;
            __builtin_amdgcn_global_load_async_to_lds_b128(
                (AS1 v4i*)(src + e), (AS3 v4i*)(spos4 + e), 0, 0);
        }
#if __has_builtin(__builtin_amdgcn_s_wait_asynccnt)
        __builtin_amdgcn_s_wait_asynccnt(0);
#else
        asm volatile("s_wait_asynccnt 0x0" ::: "memory");
#endif
#else
        #pragma unroll
        for (int k = 0; k < (NN * 2 / 4) / 256; ++k) {
            const int e = threadIdx.x + k * 256;
            spos4[e] = src[e];
        }
#endif
    }
    __syncthreads();
    const float* spos = (const float*)spos4;

    __builtin_prefetch(&velb[i0 * 2], 0, 3);   // merge phase will want these

    // A fragment (16x4 f32 -> v2f per lane per ISA layout)
    float ax = spos[(i0 + r) * 2 + 0];
    float ay = spos[(i0 + r) * 2 + 1];
    float as = ax * ax + ay * ay;
    v2f afrag;
    if (h == 0) { afrag[0] = ax; afrag[1] = ay;   }   // K=0,1
    else        { afrag[0] = as; afrag[1] = 1.0f; }   // K=2,3

    // register-resident sorted top-5 (ascending d^2)
    float t0 = 3.0e38f, t1 = 3.0e38f, t2 = 3.0e38f, t3 = 3.0e38f, t4 = 3.0e38f;
    int   n0 = -1, n1 = -1, n2 = -1, n3 = -1, n4 = -1;

    const int i = i0 + r;   // global row this lane scans

    auto consider = [&](float raw, int j) {
        float d2 = fmaxf(raw, 0.0f);
        if (j == i) d2 = __builtin_inff();   // exclude self (eye + 1e6 in ref)
        if (d2 < t4) {
            t4 = d2; n4 = j;
            if (t4 < t3) { float f = t3; t3 = t4; t4 = f; int q = n3; n3 = n4; n4 = q; }
            if (t3 < t2) { float f = t2; t2 = t3; t3 = f; int q = n2; n2 = n3; n3 = q; }
            if (t2 < t1) { float f = t1; t1 = t2; t2 = f; int q = n1; n1 = n2; n2 = q; }
            if (t1 < t0) { float f = t0; t0 = t1; t1 = f; int q = n0; n0 = n1; n1 = q; }
        }
    };

    for (int tt = 0; tt < NN / 16; ++tt) {
        const int j0 = tt * 16;

        // B fragment (4x16 f32 -> v2f per lane) from LDS
        float bx = spos[(j0 + r) * 2 + 0];
        float by = spos[(j0 + r) * 2 + 1];
        float bs = bx * bx + by * by;
        v2f bfrag;
        if (h == 0) { bfrag[0] = -2.0f * bx; bfrag[1] = -2.0f * by; } // K=0,1
        else        { bfrag[0] = 1.0f;       bfrag[1] = bs;         } // K=2,3

        v8f c = {};
        v8f d = __builtin_amdgcn_wmma_f32_16x16x4_f32(
            /*neg_a=*/false, afrag, /*neg_b=*/false, bfrag,
            /*c_mod=*/(short)0, c, /*reuse_a=*/false, /*reuse_b=*/false);

        // D layout: VGPR v -> row v+8h, col = lane%16
        #pragma unroll
        for (int v = 0; v < 8; ++v)
            tile[wv][v + 8 * h][r] = d[v];
        __builtin_amdgcn_wave_barrier();      // intra-wave LDS exchange fence

        // lane scans 8 contiguous columns of its row as two b128 loads
        const float4 q0 = *(const float4*)&tile[wv][r][8 * h + 0];
        const float4 q1 = *(const float4*)&tile[wv][r][8 * h + 4];
        __builtin_amdgcn_wave_barrier();

        const int jb = j0 + 8 * h;
        consider(q0.x, jb + 0); consider(q0.y, jb + 1);
        consider(q0.z, jb + 2); consider(q0.w, jb + 3);
        consider(q1.x, jb + 4); consider(q1.y, jb + 5);
        consider(q1.z, jb + 6); consider(q1.w, jb + 7);
        __builtin_amdgcn_wave_barrier();      // keep next stores after loads
    }

    // publish per-half lists, merge in lanes 0..15 (intra-wave exchange)
    md[wv][h][r][0] = t0; md[wv][h][r][1] = t1; md[wv][h][r][2] = t2;
    md[wv][h][r][3] = t3; md[wv][h][r][4] = t4;
    mi[wv][h][r][0] = n0; mi[wv][h][r][1] = n1; mi[wv][h][r][2] = n2;
    mi[wv][h][r][3] = n3; mi[wv][h][r][4] = n4;
    __builtin_amdgcn_wave_barrier();

    float part = 0.0f;
    if (h == 0) {
        const float vix = velb[i * 2 + 0];
        const float viy = velb[i * 2 + 1];
        int pa = 0, pb = 0;
        #pragma unroll
        for (int k = 0; k < TOPK; ++k) {
            float da = md[wv][0][r][pa], db = md[wv][1][r][pb];
            int   ia = mi[wv][0][r][pa], ib = mi[wv][1][r][pb];
            bool takeA = (da < db) || (da == db && ia < ib);
            float d2 = takeA ? da : db;
            int   j  = takeA ? ia : ib;
            if (takeA) ++pa; else ++pb;
            float dist = sqrtf(d2);
            float w = 1.0f / (dist + 1e-6f);
            float dvx = vix - velb[j * 2 + 0];
            float dvy = viy - velb[j * 2 + 1];
            part += sqrtf(dvx * dvx + dvy * dvy) * w;
        }
    }
    float s = blk_reduce(part, red);
    if (threadIdx.x == 0) atomicAdd(&ws[1], s);
}

// ---------------- boundary loss partials -> ws[2] (sum), ws[3] (count) -------
__global__ __launch_bounds__(256) void bnd_kernel(const float* __restrict__ pos,
                                                  const float* __restrict__ vel,
                                                  float* __restrict__ ws) {
    __shared__ float red[8];
    float s = 0.0f, c = 0.0f;
    for (int i = blockIdx.x * blockDim.x + threadIdx.x; i < BB * NN;
         i += gridDim.x * blockDim.x) {
        float x = pos[i * 2 + 0], y = pos[i * 2 + 1];
        bool mask = (x < 0.05f) | (x > 1.0f - 0.05f) | (y < 0.05f) | (y > 1.0f - 0.05f);
        float m0 = x, m1 = 1.0f - x, m2 = y, m3 = 1.0f - y;
        int bt = 0; float mv = m0;
        if (m1 < mv) { mv = m1; bt = 1; }     // first-index argmin semantics
        if (m2 < mv) { mv = m2; bt = 2; }
        if (m3 < mv) { mv = m3; bt = 3; }
        float nx = (bt == 0) ? -1.0f : (bt == 1) ? 1.0f : 0.0f;
        float ny = (bt == 2) ? -1.0f : (bt == 3) ? 1.0f : 0.0f;
        float nc = vel[i * 2 + 0] * nx + vel[i * 2 + 1] * ny;
        if (mask) { s += nc * nc; c += 1.0f; }
    }
    float rs = blk_reduce(s, red);
    if (threadIdx.x == 0) atomicAdd(&ws[2], rs);
    float rc = blk_reduce(c, red);
    if (threadIdx.x == 0) atomicAdd(&ws[3], rc);
}

// ---------------- obstacle loss: one block per (b,m) -------------------------
__global__ __launch_bounds__(256) void obs_kernel(const float* __restrict__ pos,
                                                  const float* __restrict__ vel,
                                                  const float* __restrict__ obst,
                                                  float* __restrict__ ws) {
    __shared__ float red[8];
    const int bm = blockIdx.x;              // 0..BB*MM-1
    const int b = bm / MM;
    const float cx = obst[bm * 3 + 0];
    const float cy = obst[bm * 3 + 1];
    const float rr = obst[bm * 3 + 2];
    float pen = 0.0f, cnt = 0.0f;
    for (int n = threadIdx.x; n < NN; n += blockDim.x) {
        float px = pos[((size_t)b * NN + n) * 2 + 0];
        float py = pos[((size_t)b * NN + n) * 2 + 1];
        float dx = px - cx, dy = py - cy;
        float d = sqrtf(dx * dx + dy * dy);
        bool near = d < rr * 2.0f;
        float inv = 1.0f / (d + 1e-6f);
        float w = expf(-(d - rr) / (rr * 0.5f));
        float proj = vel[((size_t)b * NN + n) * 2 + 0] * (dx * inv) +
                     vel[((size_t)b * NN + n) * 2 + 1] * (dy * inv);
        float neg = fmaxf(-proj, 0.0f);
        float pp = w * neg * neg;
        if (near) { pen += pp; cnt += 1.0f; }
    }
    float ps = blk_reduce(pen, red);
    if (threadIdx.x == 0) ws[8 + bm] = ps;
    float cs = blk_reduce(cnt, red);
    if (threadIdx.x == 0) ws[72 + bm] = cs;
}

// ---------------- divergence loss (deterministic hashed sample indices) ------
__global__ __launch_bounds__(512) void div_kernel(const float* __restrict__ pos,
                                                  const float* __restrict__ vel,
                                                  float* __restrict__ ws) {
    __shared__ float red[16];
    const int tid = blockIdx.x * blockDim.x + threadIdx.x;
    float acc = 0.0f;
    if (tid < BB * NDIV) {
        const int b = tid / NDIV, s = tid % NDIV;
        const uint32_t i0 = hashu((uint32_t)(s * 4 + 1)) & (NN - 1);
        const uint32_t i1 = hashu((uint32_t)(s * 4 + 2)) & (NN - 1);
        const uint32_t i2 = hashu((uint32_t)(s * 4 + 3)) & (NN - 1);
        const size_t base = (size_t)b * NN;
        float dx = pos[(base + i1) * 2 + 0] - pos[(base + i0) * 2 + 0];
        float dy = pos[(base + i2) * 2 + 1] - pos[(base + i0) * 2 + 1];
        float dvx = (vel[(base + i1) * 2 + 0] - vel[(base + i0) * 2 + 0]) / (dx + 1e-6f);
        float dvy = (vel[(base + i2) * 2 + 1] - vel[(base + i0) * 2 + 1]) / (dy + 1e-6f);
        float dsum = dvx + dvy;
        acc = dsum * dsum;
    }
    float s2 = blk_reduce(acc, red);
    if (threadIdx.x == 0) atomicAdd(&ws[4], s2);
}

// ---------------- finalize: normalize + weight -> out[0] ---------------------
__global__ void fin_kernel(const float* __restrict__ obst,
                           const float* __restrict__ ws,
                           float* __restrict__ out) {
    float vel_loss = ws[0] / (float)(BB * NN * 2);
    float cons     = ws[1] / (float)(BB * NN * TOPK);
    float bnd      = (ws[3] > 0.0f) ? ws[2] / fmaxf(ws[3], 1.0f) : 0.0f;
    float obs = 0.0f;
    for (int bm = 0; bm < BB * MM; ++bm) {
        float rr = obst[bm * 3 + 2];
        float cnt = ws[72 + bm];
        float pen = (rr > 0.0f && cnt > 0.0f) ? ws[8 + bm] / fmaxf(cnt, 1.0f) : 0.0f;
        obs += pen;
    }
    obs /= (float)(BB * MM);
    float dv = ws[4] / (float)(BB * NDIV);
    out[0] = 1.0f * vel_loss + 0.1f * cons + 0.5f * bnd + 1.0f * obs + 0.1f * dv;
}

extern "C" void kernel_launch(void* const* d_in, const int* in_sizes, int n_in,
                              void* d_out, int out_size, void* d_ws, size_t ws_size,
                              hipStream_t stream) {
    const float* pred = (const float*)d_in[0];   // (B,N,2)
    const float* targ = (const float*)d_in[1];   // (B,N,2)
    const float* pos  = (const float*)d_in[2];   // (B,N,2)
    const float* obst = (const float*)d_in[3];   // (B,M,3)
    float* out = (float*)d_out;
    float* ws  = (float*)d_ws;

    (void)hipMemsetAsync(ws, 0, 160 * sizeof(float), stream);

    vel_kernel<<<32, 256, 0, stream>>>(pred, targ, ws);
    cons_kernel<<<(BB * (NN / 16)) / 8, 256, 0, stream>>>(pos, pred, ws);
    bnd_kernel<<<16, 256, 0, stream>>>(pos, pred, ws);
    obs_kernel<<<BB * MM, 256, 0, stream>>>(pos, pred, obst, ws);
    div_kernel<<<1, 512, 0, stream>>>(pos, pred, ws);
    fin_kernel<<<1, 1, 0, stream>>>(obst, ws, out);
}